// SimpleLSTM_46377056862585
// MI455X (gfx1250) — compile-verified
//
#include <hip/hip_runtime.h>
#include <cmath>

#define HSZ 1024
#define ISZ 1024
#define BSZ 64
#define TSZ 512

typedef __attribute__((ext_vector_type(2))) float v2f;
typedef __attribute__((ext_vector_type(4))) float v4f;
typedef __attribute__((ext_vector_type(8))) float v8f;
typedef __attribute__((ext_vector_type(4))) int   v4i;

// Global-address-space pointer types (force global_load_*, never flat)
typedef const __attribute__((address_space(1))) float gfloat;
typedef const __attribute__((address_space(1))) v2f   gv2f;

// ---- CDNA5 async global->LDS path (guarded; sync fallback keeps compile safe)
#if defined(__has_builtin)
#  if __has_builtin(__builtin_amdgcn_global_load_async_to_lds_b128)
#    define HAVE_ASYNC 1
#  else
#    define HAVE_ASYNC 0
#  endif
#else
#  define HAVE_ASYNC 0
#endif

#if HAVE_ASYNC
#  if __has_builtin(__builtin_amdgcn_s_wait_asynccnt)
#    define WAIT_ASYNC(n) __builtin_amdgcn_s_wait_asynccnt(n)
#  else
#    define WAIT_ASYNC(n) asm volatile("s_wait_asynccnt %0" ::"n"(n) : "memory")
#  endif
// param1: global source (AS1, v4i*); param2: LDS dest (AS3, v4i*)
#  define ASG128(p) ((__attribute__((address_space(1))) v4i*)(const_cast<float*>(p)))
#  define ASL128(p) ((__attribute__((address_space(3))) v4i*)(p))
#  define ASYNC_CP16(gsrc, ldst) \
      __builtin_amdgcn_global_load_async_to_lds_b128(ASG128(gsrc), ASL128(ldst), 0, 0)
#endif

// ---------------------------------------------------------------------------
// Utility kernels
// ---------------------------------------------------------------------------
__global__ void zero_kernel(float* __restrict__ p, int n) {
    int i = blockIdx.x * blockDim.x + threadIdx.x;
    if (i < n) p[i] = 0.0f;
}

// src is [K=1024][N=1024] row-major; dst becomes [N][K] (coalesced writes).
__global__ void transpose_kernel(const float* __restrict__ src, float* __restrict__ dst) {
    int idx = blockIdx.x * blockDim.x + threadIdx.x;   // exactly 1024*1024 threads
    int n = idx >> 10;
    int k = idx & 1023;
    dst[(size_t)n * 1024 + k] = src[(size_t)k * 1024 + n];
}

__global__ void tail_kernel(float* __restrict__ out, const float* __restrict__ cbuf) {
    int i = blockIdx.x * blockDim.x + threadIdx.x;     // B*H threads
    if (i < BSZ * HSZ) {
        int b = i >> 10;
        int h = i & 1023;
        out[(size_t)BSZ * TSZ * HSZ + i] =
            out[(size_t)b * TSZ * HSZ + (size_t)(TSZ - 1) * HSZ + h];
        out[(size_t)BSZ * TSZ * HSZ + (size_t)BSZ * HSZ + i] = cbuf[i];
    }
}

// ---------------------------------------------------------------------------
// One LSTM timestep, fused 4 gates + cell update.
// Block = 512 threads (16 wave32). Wave w: gate g=w&3, M-quarter q=w>>2.
// Grid.x = H/16 = 64 N-tiles -> 1024 waves/step for latency hiding.
// Each wave: one 16x16 f32 accumulator, rows [q*16,q*16+16), cols [n0,n0+16),
// accumulated over K=1024 (x_t @ Wx) + K=1024 (h_{t-1} @ Wh).
// A tiles (64 rows x 64 K): double-buffered in LDS via async global->LDS;
// B tiles: prefetched to registers before the barrier (global_load_b64).
// ---------------------------------------------------------------------------
#define ASTR 68        // LDS row stride (floats): 16B-aligned rows, bank-spread
#define NBLK 32        // 2 passes x 16 K-blocks of 64

__launch_bounds__(512)
__global__ void lstm_step_kernel(const float* __restrict__ x,     // [B,T,I]
                                 const float* __restrict__ WTx,   // [4][H][I] (N-major)
                                 const float* __restrict__ WTh,   // [4][H][H] (N-major)
                                 const float* __restrict__ b_i,
                                 const float* __restrict__ b_f,
                                 const float* __restrict__ b_c,
                                 const float* __restrict__ b_o,
                                 const float* __restrict__ hprev, // h_{t-1} base
                                 long hstride,                    // row stride of hprev
                                 float* __restrict__ cbuf,        // [B,H] cell state
                                 float* __restrict__ out,         // d_out [B,T,H]
                                 int t) {
    __shared__ float smemA[2][64 * ASTR];   // double-buffered A staging
    __shared__ float smemP[4 * 64 * 16];    // gate pre-activation staging

    const int tid  = threadIdx.x;
    const int wave = tid >> 5;
    const int lane = tid & 31;
    const int g    = wave & 3;          // gate: 0=i 1=f 2=c 3=o
    const int q    = wave >> 2;         // M quarter: rows [q*16, q*16+16)
    const int n0   = blockIdx.x * 16;
    const int nl   = lane & 15;         // N (B/C/D) or M-in-tile (A) index
    const int kb   = (lane >> 4) * 2;   // K sub-offset per ISA 16x4/4x16 striping

    // A-tile sources: pass 0 = x_t rows, pass 1 = h_{t-1} rows
    const float* srcs[2];
    long         sstr[2];
    srcs[0] = x + (size_t)t * ISZ;  sstr[0] = (long)TSZ * ISZ;
    srcs[1] = hprev;                sstr[1] = hstride;

    // B row pointers, pinned to the global address space (no FLAT fallback)
    gfloat* wxrow = (gfloat*)(WTx + (size_t)g * HSZ * ISZ + (size_t)(n0 + nl) * ISZ + kb);
    gfloat* whrow = (gfloat*)(WTh + (size_t)g * HSZ * HSZ + (size_t)(n0 + nl) * HSZ + kb);

    // staging map: 64 rows x 64 floats, 8 floats/thread (2 x b128)
    const int sr = tid >> 3;            // 0..63
    const int sc = (tid & 7) * 8;       // 0,8,..,56

    v8f acc = {};

    // ---- prologue: stage block 0
    {
        const float* s = srcs[0] + (size_t)sr * sstr[0] + sc;
        float* d = &smemA[0][sr * ASTR + sc];
#if HAVE_ASYNC
        ASYNC_CP16(s, d);
        ASYNC_CP16(s + 4, d + 4);
#else
        *(v4f*)d = *(const v4f*)s;
        *(v4f*)(d + 4) = *(const v4f*)(s + 4);
#endif
    }

    for (int j = 0; j < NBLK; ++j) {
        const int cur = j & 1;

        // ---- prefetch current block's B tile into registers (global_load_b64 x16)
        gfloat* wrow = ((j < 16) ? wxrow : whrow) + (j & 15) * 64;
        v2f breg[16];
#pragma unroll
        for (int u = 0; u < 16; ++u)
            breg[u] = *(gv2f*)(wrow + u * 4);

        // ---- issue async stage of next block into the other buffer
        if (j + 1 < NBLK) {
            const int jp   = j + 1;
            const int pasn = jp >> 4;
            const int k0n  = (jp & 15) * 64;
            const float* s = srcs[pasn] + (size_t)sr * sstr[pasn] + k0n + sc;
            float* d = &smemA[cur ^ 1][sr * ASTR + sc];
#if HAVE_ASYNC
            ASYNC_CP16(s, d);
            ASYNC_CP16(s + 4, d + 4);
#else
            *(v4f*)d = *(const v4f*)s;
            *(v4f*)(d + 4) = *(const v4f*)(s + 4);
#endif
        }
#if HAVE_ASYNC
        if (j + 1 < NBLK) { WAIT_ASYNC(2); } else { WAIT_ASYNC(0); }
#endif
        __syncthreads();   // current A tile visible to all waves

        // ---- 16 K-chunks of 4: A from LDS, B from registers
        const float* arow = &smemA[cur][(q * 16 + nl) * ASTR + kb];
#pragma unroll
        for (int kk = 0; kk < 16; ++kk) {
            v2f a = *(const v2f*)(arow + kk * 4);    // A: (M=nl, K=4*kk+kb..)
            acc = __builtin_amdgcn_wmma_f32_16x16x4_f32(
                false, a, false, breg[kk], (short)0, acc, false, false);
        }
        __syncthreads();   // tile consumed; buffer may be overwritten in 2 blocks
    }

    // ---- stage gate pre-activations: pre[g][m][nl]
    {
        const int hi = lane >> 4;
#pragma unroll
        for (int r = 0; r < 8; ++r) {
            int ml = r + 8 * hi;                     // C/D layout: M = r + 8*hi
            smemP[(g * 64 + q * 16 + ml) * 16 + nl] = acc[r];
        }
    }
    __syncthreads();

    // ---- elementwise LSTM cell update over the 64x16 tile (2 elems/thread)
#pragma unroll
    for (int j = 0; j < 2; ++j) {
        int idx = tid + j * 512;          // 0..1023
        int m = idx >> 4;
        int nn = idx & 15;
        int n = n0 + nn;
        float pi = smemP[(0 * 64 + m) * 16 + nn] + b_i[n];
        float pf = smemP[(1 * 64 + m) * 16 + nn] + b_f[n];
        float pc = smemP[(2 * 64 + m) * 16 + nn] + b_c[n];
        float po = smemP[(3 * 64 + m) * 16 + nn] + b_o[n];
        float ig = 1.0f / (1.0f + __expf(-pi));
        float fg = 1.0f / (1.0f + __expf(-pf));
        float cg = tanhf(pc);
        float og = 1.0f / (1.0f + __expf(-po));
        float cold = cbuf[m * HSZ + n];
        float cnew = fg * cold + ig * cg;
        cbuf[m * HSZ + n] = cnew;
        out[(size_t)m * TSZ * HSZ + (size_t)t * HSZ + n] = og * tanhf(cnew);
    }
}

// ---------------------------------------------------------------------------
// Host side
// ---------------------------------------------------------------------------
extern "C" void kernel_launch(void* const* d_in, const int* in_sizes, int n_in,
                              void* d_out, int out_size, void* d_ws, size_t ws_size,
                              hipStream_t stream) {
    (void)in_sizes; (void)n_in; (void)out_size; (void)ws_size;

    const float* x  = (const float*)d_in[0];
    const float* bi = (const float*)d_in[3];
    const float* bf = (const float*)d_in[6];
    const float* bc = (const float*)d_in[9];
    const float* bo = (const float*)d_in[12];
    const float* wx[4] = { (const float*)d_in[1], (const float*)d_in[4],
                           (const float*)d_in[7], (const float*)d_in[10] };
    const float* wh[4] = { (const float*)d_in[2], (const float*)d_in[5],
                           (const float*)d_in[8], (const float*)d_in[11] };

    // Workspace (floats): WTx[4][H][I] | WTh[4][H][H] | cbuf[B*H] | zbuf[H]
    float* wsf  = (float*)d_ws;
    float* WTx  = wsf;
    float* WTh  = WTx + (size_t)4 * HSZ * ISZ;
    float* cbuf = WTh + (size_t)4 * HSZ * HSZ;
    float* zbuf = cbuf + (size_t)BSZ * HSZ;

    zero_kernel<<<(BSZ * HSZ + HSZ + 255) / 256, 256, 0, stream>>>(cbuf, BSZ * HSZ + HSZ);

    for (int g = 0; g < 4; ++g) {
        transpose_kernel<<<4096, 256, 0, stream>>>(wx[g], WTx + (size_t)g * HSZ * ISZ);
        transpose_kernel<<<4096, 256, 0, stream>>>(wh[g], WTh + (size_t)g * HSZ * HSZ);
    }

    float* out = (float*)d_out;
    for (int t = 0; t < TSZ; ++t) {
        const float* hprev = (t == 0) ? zbuf : out + (size_t)(t - 1) * HSZ;
        long hstride = (t == 0) ? 0 : (long)TSZ * HSZ;
        lstm_step_kernel<<<HSZ / 16, 512, 0, stream>>>(
            x, WTx, WTh, bi, bf, bc, bo, hprev, hstride, cbuf, out, t);
    }

    tail_kernel<<<(BSZ * HSZ + 255) / 256, 256, 0, stream>>>(out, cbuf);
}